// MultiHopGNN_83502754169571
// MI455X (gfx1250) — compile-verified
//
#include <hip/hip_runtime.h>

typedef __attribute__((ext_vector_type(2))) float v2f;
typedef __attribute__((ext_vector_type(8))) float v8f;

#define N_NODES 50000
#define N_EDGES 800000
#define IN_FT   128
#define HID_FT  64
#define EMB_FT  128
#define CAT_FT  256   // 4 * HID_FT
#define NEG_SLOP 0.2f

__device__ __forceinline__ float lrelu(float v) {
    // for v>=0: v > 0.2v ; for v<0: 0.2v > v  -> max implements leaky relu
    return fmaxf(v, NEG_SLOP * v);
}

// ---------------------------------------------------------------- degree prep
__global__ void k_init_deg(float* deg) {
    int i = blockIdx.x * blockDim.x + threadIdx.x;
    if (i < N_NODES) deg[i] = 1.0f;                 // self-loop contributes 1
}

__global__ void k_edge_deg(const int* __restrict__ ei, const float* __restrict__ ew,
                           float* __restrict__ deg) {
    int e = blockIdx.x * blockDim.x + threadIdx.x;
    if (e < N_EDGES) unsafeAtomicAdd(&deg[ei[N_EDGES + e]], ew[e]);
}

__global__ void k_deg_to_dinv(float* deg) {
    int i = blockIdx.x * blockDim.x + threadIdx.x;
    if (i < N_NODES) {
        float d = deg[i];
        deg[i] = (d > 0.f) ? rsqrtf(d) : 0.f;
    }
}

__global__ void k_zero(float* __restrict__ p, int n) {
    int i = blockIdx.x * blockDim.x + threadIdx.x;
    if (i < n) p[i] = 0.f;
}

// ---------------------------------------------------------------- WMMA GEMM
// C[16 x F] tile per block. blockDim.x = 2*F (one wave per 16-col tile).
// A tile (16 x K) staged through LDS; B (K x F) read from global (hot in L2).
// Requires: N rows multiple of 16 (50000 = 3125*16), K multiple of 4, K <= 256.
__global__ void k_gemm_wmma(const float* __restrict__ A, int lda, int K,
                            const float* __restrict__ W, int F,
                            const float* __restrict__ bias,
                            float* __restrict__ C, int ldc) {
    __shared__ float As[16 * 256];                  // 16 KB max (K<=256)

    const int m0   = blockIdx.x * 16;
    const int tid  = threadIdx.x;
    const int nthr = blockDim.x;

    // cooperative stage of the 16 x K A-tile
    for (int idx = tid; idx < 16 * K; idx += nthr) {
        int r = idx / K, c = idx - r * K;
        As[r * K + c] = A[(size_t)(m0 + r) * lda + c];
    }
    __syncthreads();

    const int lane = tid & 31;
    const int wave = tid >> 5;
    const int col0 = wave * 16;
    const int l15  = lane & 15;          // M row (A frag) / N col (B frag)
    const int kb   = (lane >> 4) * 2;    // K sub-pair selected by lane half

    v8f acc = {};
    for (int k = 0; k < K; k += 4) {
        v2f a, b;
        // A fragment: 16x4, lane holds A[m][k+kb], A[m][k+kb+1]
        a.x = As[l15 * K + k + kb];
        a.y = As[l15 * K + k + kb + 1];
        // B fragment: 4x16, lane holds B[k+kb][n], B[k+kb+1][n]
        const float* Bp = W + (size_t)(k + kb) * F + col0 + l15;
        b.x = Bp[0];
        b.y = Bp[F];
        acc = __builtin_amdgcn_wmma_f32_16x16x4_f32(
            /*neg_a=*/false, a, /*neg_b=*/false, b,
            /*c_mod=*/(short)0, acc, /*reuse_a=*/false, /*reuse_b=*/false);
    }

    // C/D layout: VGPR r -> M = r + 8*(lane>=16), N = lane&15
    const int   mb = (lane >> 4) * 8;
    const float bv = bias ? bias[col0 + l15] : 0.f;
#pragma unroll
    for (int r = 0; r < 8; ++r) {
        C[(size_t)(m0 + mb + r) * ldc + col0 + l15] = acc[r] + bv;
    }
}

// ---------------------------------------------------------------- edge scatter
// 16 lanes per edge; each lane moves one float4 (4 features) of the 64-wide row.
__global__ void k_scatter(const int* __restrict__ ei, const float* __restrict__ ew,
                          const float* __restrict__ dinv,
                          const float* __restrict__ H, float* __restrict__ AGG,
                          int use_norm) {
    int gid = blockIdx.x * blockDim.x + threadIdx.x;
    int e   = gid >> 4;
    if (e >= N_EDGES) return;
    int c = gid & 15;

    int   s = ei[e];
    int   d = ei[N_EDGES + e];
    float w = ew[e];
    if (use_norm) w *= dinv[s] * dinv[d];

    const float4 hv = ((const float4*)H)[(size_t)s * 16 + c];
    float* dst = AGG + (size_t)d * HID_FT + c * 4;
    unsafeAtomicAdd(dst + 0, w * hv.x);
    unsafeAtomicAdd(dst + 1, w * hv.y);
    unsafeAtomicAdd(dst + 2, w * hv.z);
    unsafeAtomicAdd(dst + 3, w * hv.w);
}

// ---------------------------------------------------------------- epilogue
// h = lrelu(AGG + [dinv^2 * H] + bias) written into CAT column block col_off.
__global__ void k_combine(const float* __restrict__ AGG, const float* __restrict__ H,
                          const float* __restrict__ dinv, const float* __restrict__ bias,
                          float* __restrict__ CAT, int col_off, int use_self) {
    int gid = blockIdx.x * blockDim.x + threadIdx.x;
    if (gid >= N_NODES * HID_FT) return;
    int i = gid >> 6;   // node
    int f = gid & 63;   // feature

    float v = AGG[gid] + bias[f];
    if (use_self) {
        float di = dinv[i];
        v += di * di * H[gid];
    }
    CAT[(size_t)i * CAT_FT + col_off + f] = lrelu(v);
}

// ---------------------------------------------------------------- launcher
extern "C" void kernel_launch(void* const* d_in, const int* in_sizes, int n_in,
                              void* d_out, int out_size, void* d_ws, size_t ws_size,
                              hipStream_t stream) {
    const float* x  = (const float*)d_in[0];
    const int*   ei = (const int*)  d_in[1];   // [2, E]: row0 = src, row1 = dst
    const float* ew = (const float*)d_in[2];
    const float* W1 = (const float*)d_in[3];  const float* b1 = (const float*)d_in[4];
    const float* W2 = (const float*)d_in[5];  const float* b2 = (const float*)d_in[6];
    const float* W3 = (const float*)d_in[7];  const float* b3 = (const float*)d_in[8];
    const float* W4 = (const float*)d_in[9];  const float* b4 = (const float*)d_in[10];
    const float* Wh = (const float*)d_in[11]; const float* bh = (const float*)d_in[12];
    float* out = (float*)d_out;

    // workspace layout (floats)
    float* ws   = (float*)d_ws;
    float* dinv = ws;                                   // N (deg -> dinv in place)
    float* H    = ws + 50176;                           // N*64
    float* AGG  = H  + (size_t)N_NODES * HID_FT;        // N*64
    float* CAT  = AGG + (size_t)N_NODES * HID_FT;       // N*256

    const int TB = 256;
    const int gN   = (N_NODES + TB - 1) / TB;
    const int gE   = (N_EDGES + TB - 1) / TB;
    const int gNF  = (N_NODES * HID_FT + TB - 1) / TB;
    const int gSC  = (N_EDGES * 16 + TB - 1) / TB;
    const int gMM  = N_NODES / 16;                      // 3125 row tiles, exact

    // degree / symmetric norm
    k_init_deg   <<<gN, TB, 0, stream>>>(dinv);
    k_edge_deg   <<<gE, TB, 0, stream>>>(ei, ew, dinv);
    k_deg_to_dinv<<<gN, TB, 0, stream>>>(dinv);

    // ---- layer 1: no-norm GCN
    k_gemm_wmma<<<gMM, 2 * HID_FT, 0, stream>>>(x, IN_FT, IN_FT, W1, HID_FT, nullptr, H, HID_FT);
    k_zero     <<<gNF, TB, 0, stream>>>(AGG, N_NODES * HID_FT);
    k_scatter  <<<gSC, TB, 0, stream>>>(ei, ew, dinv, H, AGG, /*use_norm=*/0);
    k_combine  <<<gNF, TB, 0, stream>>>(AGG, H, dinv, b1, CAT, /*col_off=*/0, /*use_self=*/0);

    // ---- layers 2..4: normalized GCN, input is previous CAT column block
    const float* Ws[3] = {W2, W3, W4};
    const float* bs[3] = {b2, b3, b4};
    for (int l = 0; l < 3; ++l) {
        k_gemm_wmma<<<gMM, 2 * HID_FT, 0, stream>>>(CAT + (size_t)l * HID_FT, CAT_FT,
                                                    HID_FT, Ws[l], HID_FT, nullptr, H, HID_FT);
        k_zero     <<<gNF, TB, 0, stream>>>(AGG, N_NODES * HID_FT);
        k_scatter  <<<gSC, TB, 0, stream>>>(ei, ew, dinv, H, AGG, /*use_norm=*/1);
        k_combine  <<<gNF, TB, 0, stream>>>(AGG, H, dinv, bs[l], CAT,
                                            /*col_off=*/(l + 1) * HID_FT, /*use_self=*/1);
    }

    // ---- final projection: out = CAT @ Wh + bh   (K=256, F=128, 8 waves/block)
    k_gemm_wmma<<<gMM, 2 * EMB_FT, 0, stream>>>(CAT, CAT_FT, CAT_FT, Wh, EMB_FT, bh, out, EMB_FT);
}